// GCNEncoder_6923487282673
// MI455X (gfx1250) — compile-verified
//
#include <hip/hip_runtime.h>
#include <hip/hip_bf16.h>

typedef __attribute__((ext_vector_type(2))) float v2f;
typedef __attribute__((ext_vector_type(8))) float v8f;
typedef __attribute__((ext_vector_type(4))) unsigned int u32x4;
typedef __attribute__((ext_vector_type(8))) int i32x8;
typedef __attribute__((ext_vector_type(4))) int i32x4;

#define HID 256
#define BN_EPS 1e-5f

// ---------------------------------------------------------------------------
// generic zero
__global__ void gcn_zero_kernel(float* __restrict__ p, long long n) {
    long long i = (long long)blockIdx.x * blockDim.x + threadIdx.x;
    long long stride = (long long)gridDim.x * blockDim.x;
    for (; i < n; i += stride) p[i] = 0.0f;
}

// degree: deg[dst] += 1 for every edge and every self loop
__global__ void gcn_degree_kernel(const int* __restrict__ dstI, float* __restrict__ deg,
                                  int nEdges, int nNodes) {
    int i = blockIdx.x * blockDim.x + threadIdx.x;
    int total = nEdges + nNodes;
    if (i >= total) return;
    int d = (i < nEdges) ? dstI[i] : (i - nEdges);
    atomicAdd(&deg[d], 1.0f);
}

// dis = deg > 0 ? rsqrt(deg) : 0   (in place)
__global__ void gcn_rsqrt_kernel(float* __restrict__ deg, int n) {
    int i = blockIdx.x * blockDim.x + threadIdx.x;
    if (i >= n) return;
    float d = deg[i];
    deg[i] = (d > 0.0f) ? rsqrtf(d) : 0.0f;
}

// ---------------------------------------------------------------------------
// TDM: stage a 16 x K f32 tile (row stride K) from global into LDS.
// D# layout per CDNA5 ISA ch.8: group0 = {flags, lds_addr, global_addr, type},
// group1 = {data_size, tensor dims/strides, tile dims}. Groups 2/3 zero (2D).
template <int K>
__device__ __forceinline__ void tdm_load_tile_to_lds(const float* gsrc, unsigned ldsBase) {
    unsigned long long ga = (unsigned long long)(const void*)gsrc;

    u32x4 g0;
    g0[0] = 1u;                                              // count=1, user descriptor
    g0[1] = ldsBase;                                         // lds_addr (bytes)
    g0[2] = (unsigned)(ga & 0xFFFFFFFFu);                    // global_addr[31:0]
    g0[3] = (unsigned)((ga >> 32) & 0x01FFFFFFu)             // global_addr[56:32]
          | 0x80000000u;                                     // type = 2 ("image")

    i32x8 g1;
    g1[0] = 0x20000;                 // workgroup_mask=0, data_size=2 (4 bytes)
    g1[1] = (K & 0xFFFF) << 16;      // tensor_dim0[15:0] = K   (bits 79:48 lo half)
    g1[2] = 16 << 16;                // tensor_dim0 hi = 0 | tensor_dim1 lo = 16
    g1[3] = (K & 0xFFFF) << 16;      // tensor_dim1 hi = 0 | tile_dim0 = K
    g1[4] = 16;                      // tile_dim1 = 16, tile_dim2 = 0
    g1[5] = K;                       // tensor_dim0_stride lo = K (elements)
    g1[6] = 0;                       // stride hi / dim1_stride lo
    g1[7] = 0;

    i32x4 g2 = {0, 0, 0, 0};
    i32x4 g3 = {0, 0, 0, 0};

#if __clang_major__ >= 23
    i32x8 g4 = {0, 0, 0, 0, 0, 0, 0, 0};
    __builtin_amdgcn_tensor_load_to_lds(g0, g1, g2, g3, g4, 0);
#else
    __builtin_amdgcn_tensor_load_to_lds(g0, g1, g2, g3, 0);
#endif
}

// ---------------------------------------------------------------------------
// H[3125*16 x 256] = X[.. x K] @ W[K x 256] using V_WMMA_F32_16X16X4_F32.
// blockDim = 128 (4 waves). Block owns one 16-row M tile; wave w owns columns
// [64w, 64w+64) as four 16x16 tiles. A tile staged in LDS by the TDM, shared
// by all 4 waves; dependency enforced via TENSORcnt + workgroup barrier.
template <int K>
__global__ void __launch_bounds__(128)
gcn_gemm_wmma_kernel(const float* __restrict__ X, const float* __restrict__ W,
                     float* __restrict__ H) {
    __shared__ float As[16 * K];

    const int tileM = blockIdx.x * 16;
    const int tid = threadIdx.x;

    // TDM async tile load: wave 0 issues (TDM ignores EXEC; avoid 4x issue),
    // waits on TENSORcnt, then the workgroup barrier publishes LDS to all.
    if (tid < 32) {
        unsigned ldsBase = (unsigned)(unsigned long long)(void*)As;
        tdm_load_tile_to_lds<K>(X + (long long)tileM * K, ldsBase);
        __builtin_amdgcn_s_wait_tensorcnt(0);
    }
    __syncthreads();

    const int wave   = tid >> 5;        // 0..3
    const int lane   = tid & 31;
    const int lhalf  = lane >> 4;       // 0 -> K pair {k,k+1}; 1 -> {k+2,k+3}
    const int l16    = lane & 15;
    const int colBase = wave * 64;

    v8f c0 = {}, c1 = {}, c2 = {}, c3 = {};

    for (int k = 0; k < K; k += 4) {
        const int kk = k + 2 * lhalf;
        v2f a;
        a.x = As[l16 * K + kk];
        a.y = As[l16 * K + kk + 1];

        const float* Wk  = W + (long long)kk * HID;
        const float* Wk1 = Wk + HID;
        // prefetch B rows a few k-steps ahead (speculative; OOB is dropped)
        __builtin_prefetch(Wk + 8 * HID + colBase + l16, 0, 0);

        v2f b0; b0.x = Wk[colBase +  0 + l16]; b0.y = Wk1[colBase +  0 + l16];
        v2f b1; b1.x = Wk[colBase + 16 + l16]; b1.y = Wk1[colBase + 16 + l16];
        v2f b2; b2.x = Wk[colBase + 32 + l16]; b2.y = Wk1[colBase + 32 + l16];
        v2f b3; b3.x = Wk[colBase + 48 + l16]; b3.y = Wk1[colBase + 48 + l16];

        c0 = __builtin_amdgcn_wmma_f32_16x16x4_f32(false, a, false, b0, (short)0, c0, false, false);
        c1 = __builtin_amdgcn_wmma_f32_16x16x4_f32(false, a, false, b1, (short)0, c1, false, false);
        c2 = __builtin_amdgcn_wmma_f32_16x16x4_f32(false, a, false, b2, (short)0, c2, false, false);
        c3 = __builtin_amdgcn_wmma_f32_16x16x4_f32(false, a, false, b3, (short)0, c3, false, false);
    }

    // C/D layout: VGPR v, lanes 0-15 -> row v, lanes 16-31 -> row v+8; col = lane&15
    #pragma unroll
    for (int v = 0; v < 8; ++v) {
        const int row = tileM + v + 8 * lhalf;
        float* out = H + (long long)row * HID + colBase + l16;
        out[ 0] = c0[v];
        out[16] = c1[v];
        out[32] = c2[v];
        out[48] = c3[v];
    }
}

// ---------------------------------------------------------------------------
// one wave per (edge | self-loop): agg[dst] += h[src] * dis[src]*dis[dst]
__global__ void gcn_aggregate_kernel(const float* __restrict__ H,
                                     const int* __restrict__ srcI,
                                     const int* __restrict__ dstI,
                                     const float* __restrict__ dis,
                                     float* __restrict__ agg,
                                     int nEdges, int nNodes) {
    int wid  = (blockIdx.x * blockDim.x + threadIdx.x) >> 5;
    int lane = threadIdx.x & 31;
    int total = nEdges + nNodes;
    if (wid >= total) return;
    int s, d;
    if (wid < nEdges) { s = srcI[wid]; d = dstI[wid]; }
    else              { s = d = wid - nEdges; }
    const float norm = dis[s] * dis[d];
    const float* hs = H + (long long)s * HID;
    float* ad = agg + (long long)d * HID;
    #pragma unroll
    for (int i = lane; i < HID; i += 32)
        atomicAdd(&ad[i], hs[i] * norm);
}

// out = relu(gamma*(agg + b - mean)*rsqrt(var+eps) + beta)
__global__ void gcn_bias_bn_relu_kernel(const float* __restrict__ agg,
                                        const float* __restrict__ b,
                                        const float* __restrict__ gamma,
                                        const float* __restrict__ beta,
                                        const float* __restrict__ mean,
                                        const float* __restrict__ var,
                                        float* __restrict__ out, long long n) {
    long long i = (long long)blockIdx.x * blockDim.x + threadIdx.x;
    if (i >= n) return;
    int c = (int)(i & (HID - 1));
    float h = agg[i] + b[c];
    float y = gamma[c] * (h - mean[c]) * rsqrtf(var[c] + BN_EPS) + beta[c];
    out[i] = fmaxf(y, 0.0f);
}

// one wave per node: sums[batch[n]] += h[n]; cnt[batch[n]] += 1
__global__ void gcn_pool_accum_kernel(const float* __restrict__ H,
                                      const int* __restrict__ batch,
                                      float* __restrict__ sums,
                                      float* __restrict__ cnt, int nNodes) {
    int wid  = (blockIdx.x * blockDim.x + threadIdx.x) >> 5;
    int lane = threadIdx.x & 31;
    if (wid >= nNodes) return;
    int g = batch[wid];
    const float* hs = H + (long long)wid * HID;
    float* sg = sums + (long long)g * HID;
    #pragma unroll
    for (int i = lane; i < HID; i += 32)
        atomicAdd(&sg[i], hs[i]);
    if (lane == 0) atomicAdd(&cnt[g], 1.0f);
}

__global__ void gcn_pool_div_kernel(const float* __restrict__ sums,
                                    const float* __restrict__ cnt,
                                    float* __restrict__ out, int n) {
    int i = blockIdx.x * blockDim.x + threadIdx.x;
    if (i >= n) return;
    int g = i >> 8;  // / HID
    out[i] = sums[i] / fmaxf(cnt[g], 1.0f);
}

// ---------------------------------------------------------------------------
extern "C" void kernel_launch(void* const* d_in, const int* in_sizes, int n_in,
                              void* d_out, int out_size, void* d_ws, size_t ws_size,
                              hipStream_t stream) {
    const float* x      = (const float*)d_in[0];
    const int*   edges  = (const int*)d_in[1];
    const int*   batch  = (const int*)d_in[2];
    const float* W1     = (const float*)d_in[3];
    const float* b1     = (const float*)d_in[4];
    const float* gamma1 = (const float*)d_in[5];
    const float* beta1  = (const float*)d_in[6];
    const float* mean1  = (const float*)d_in[7];
    const float* var1   = (const float*)d_in[8];
    const float* W2     = (const float*)d_in[9];
    const float* b2     = (const float*)d_in[10];
    const float* gamma2 = (const float*)d_in[11];
    const float* beta2  = (const float*)d_in[12];
    const float* mean2  = (const float*)d_in[13];
    const float* var2   = (const float*)d_in[14];

    const int N = in_sizes[2];          // 50000 nodes
    const int E = in_sizes[1] / 2;      // 1.6M edges
    const int G = out_size / HID;       // 128 graphs
    const int* srcI = edges;
    const int* dstI = edges + E;

    // workspace layout (16B aligned slabs)
    char* ws = (char*)d_ws;
    float* dis  = (float*)ws;                               // N
    size_t off  = ((size_t)N * 4 + 255) & ~(size_t)255;
    float* bufA = (float*)(ws + off);                       // N x 256
    off += (size_t)N * HID * 4;
    float* bufB = (float*)(ws + off);                       // N x 256
    off += (size_t)N * HID * 4;
    float* sums = (float*)(ws + off);                       // G x 256
    off += (size_t)G * HID * 4;
    float* cnt  = (float*)(ws + off);                       // G

    const long long NH = (long long)N * HID;
    const int zeroBlocks = 1024;
    const int totEdge = E + N;                              // edges + self loops
    const int aggBlocks  = (totEdge + 7) / 8;               // 8 waves / 256-thr block
    const int poolBlocks = (N + 7) / 8;
    const int mTiles = N / 16;                              // 3125

    // ---- normalization: deg -> dis = deg^-1/2 ----
    gcn_zero_kernel<<<zeroBlocks, 256, 0, stream>>>(dis, N);
    gcn_degree_kernel<<<(totEdge + 255) / 256, 256, 0, stream>>>(dstI, dis, E, N);
    gcn_rsqrt_kernel<<<(N + 255) / 256, 256, 0, stream>>>(dis, N);

    // ---- layer 1 ----
    gcn_gemm_wmma_kernel<128><<<mTiles, 128, 0, stream>>>(x, W1, bufA);     // h1
    gcn_zero_kernel<<<zeroBlocks, 256, 0, stream>>>(bufB, NH);
    gcn_aggregate_kernel<<<aggBlocks, 256, 0, stream>>>(bufA, srcI, dstI, dis, bufB, E, N);
    gcn_bias_bn_relu_kernel<<<(int)((NH + 255) / 256), 256, 0, stream>>>(
        bufB, b1, gamma1, beta1, mean1, var1, bufA, NH);                    // out1

    // ---- layer 2 ----
    gcn_gemm_wmma_kernel<256><<<mTiles, 128, 0, stream>>>(bufA, W2, bufB);  // h2
    gcn_zero_kernel<<<zeroBlocks, 256, 0, stream>>>(bufA, NH);
    gcn_aggregate_kernel<<<aggBlocks, 256, 0, stream>>>(bufB, srcI, dstI, dis, bufA, E, N);
    gcn_bias_bn_relu_kernel<<<(int)((NH + 255) / 256), 256, 0, stream>>>(
        bufA, b2, gamma2, beta2, mean2, var2, bufB, NH);                    // out2

    // ---- global mean pool ----
    gcn_zero_kernel<<<64, 256, 0, stream>>>(sums, (long long)G * HID);
    gcn_zero_kernel<<<1, 256, 0, stream>>>(cnt, G);
    gcn_pool_accum_kernel<<<poolBlocks, 256, 0, stream>>>(bufB, batch, sums, cnt, N);
    gcn_pool_div_kernel<<<(G * HID + 255) / 256, 256, 0, stream>>>(
        sums, cnt, (float*)d_out, G * HID);
}